// Uni_CrossEntropyLoss2d_12180527252165
// MI455X (gfx1250) — compile-verified
//
#include <hip/hip_runtime.h>
#include <math.h>

typedef __attribute__((ext_vector_type(2))) float v2f;
typedef __attribute__((ext_vector_type(8))) float v8f;

#define C_UNI      64
#define N_ORI_     20
#define HW_        (512*512)
#define NPIX_      (4*HW_)
#define IGNORE_IDX 255

#define BLK2  2048
#define THR2  256
#define WPB   8                    // waves per block (wave32)
#define NWAVE (BLK2*WPB)           // 16384 waves
#define NTILE (NPIX_/16)           // 65536 16-pixel tiles
#define TPW   (NTILE/NWAVE)        // 4 tiles per wave

// ---------------------------------------------------------------------------
// Main fused kernel: one streaming pass over inputs.
// Each wave owns a tile of 16 consecutive pixels (never crosses a batch image
// since H*W % 16 == 0). Per lane we cache 32 channel values in registers
// (lanes 0-15 channels {4k,4k+1}, lanes 16-31 channels {4k+2,4k+3}), take a
// per-pixel max across the two half-waves, exponentiate in registers, then do
// the 64->20 segment-sum as 16 accumulating V_WMMA_F32_16X16X4_F32 steps per
// accumulator (groups 0-15 in d0, groups 16-19 in d1).
// ---------------------------------------------------------------------------
__global__ void __launch_bounds__(THR2)
uce2d_main_kernel(const float* __restrict__ in, const int* __restrict__ tgt,
                  const int* __restrict__ gid,
                  float* __restrict__ psum, float* __restrict__ pcnt) {
  __shared__ int   s_gid[C_UNI];
  __shared__ float s_ori[WPB][16][N_ORI_];
  __shared__ float s_red[THR2];

  const int tid = threadIdx.x;
  if (tid < C_UNI) s_gid[tid] = gid[tid];
  __syncthreads();

  const int lane  = tid & 31;
  const int wave  = tid >> 5;
  const int gwave = blockIdx.x * WPB + wave;
  const int half  = lane >> 4;          // 0: channels 4k+{0,1}, 1: 4k+{2,3}
  const int col   = lane & 15;          // pixel-in-tile (A rows) / B columns

  // Precompute 0/1 selection-matrix bits for all 16 K-chunks (tile-invariant).
  unsigned m0x = 0, m0y = 0, m1x = 0, m1y = 0;
  #pragma unroll
  for (int k = 0; k < 16; ++k) {
    const int gA = s_gid[4*k + 2*half];
    const int gB = s_gid[4*k + 2*half + 1];
    m0x |= (unsigned)(gA == col)        << k;
    m0y |= (unsigned)(gB == col)        << k;
    m1x |= (unsigned)(gA == col + 16)   << k;
    m1y |= (unsigned)(gB == col + 16)   << k;
  }

  float lsum = 0.f;
  float lcnt = 0.f;

  for (int it = 0; it < TPW; ++it) {
    const int tile  = it * NWAVE + gwave;
    const int P0    = tile * 16;                 // first pixel of tile
    const int b     = P0 / HW_;                  // batch image
    const int hw    = P0 - b * HW_ + col;        // this lane's pixel offset
    const int cbase = b * C_UNI * HW_ + hw;      // channel-0 address (fits i32)

    // ---- load 32 channel values / lane (read-once stream: non-temporal) ----
    float xv[32];
    #pragma unroll
    for (int k = 0; k < 16; ++k) {
      const int cA = 4*k + 2*half;
      xv[2*k]   = __builtin_nontemporal_load(in + cbase + cA       * HW_);
      xv[2*k+1] = __builtin_nontemporal_load(in + cbase + (cA + 1) * HW_);
    }

    // ---- per-pixel max across all 64 channels (shift cancels analytically,
    //      so this is tolerance-equivalent to the reference's global max) ----
    float m = xv[0];
    #pragma unroll
    for (int i = 1; i < 32; ++i) m = fmaxf(m, xv[i]);
    m = fmaxf(m, __shfl_xor(m, 16, 32));         // combine the two half-waves

    // ---- exp in registers + per-lane partial of the denominator ----
    float tpart = 0.f;
    #pragma unroll
    for (int i = 0; i < 32; ++i) { xv[i] = __expf(xv[i] - m); tpart += xv[i]; }

    // ---- segment-sum via f32 WMMA: ori = E(16x64) @ Sel(64x20) ----
    v8f d0 = {};   // origin classes 0..15
    v8f d1 = {};   // origin classes 16..19 (cols 4..15 stay zero)
    #pragma unroll
    for (int k = 0; k < 16; ++k) {
      v2f a  = { xv[2*k], xv[2*k+1] };                       // A 16x4 slice
      v2f b0 = { (float)((m0x >> k) & 1u), (float)((m0y >> k) & 1u) };
      v2f b1 = { (float)((m1x >> k) & 1u), (float)((m1y >> k) & 1u) };
      d0 = __builtin_amdgcn_wmma_f32_16x16x4_f32(false, a, false, b0,
                                                 (short)0, d0, false, false);
      d1 = __builtin_amdgcn_wmma_f32_16x16x4_f32(false, a, false, b1,
                                                 (short)0, d1, false, false);
    }

    // denominator for pixel `col` = both half-wave partials
    const float total = __shfl(tpart, col, 32) + __shfl(tpart, col + 16, 32);

    // ---- stage D (16x20) through LDS to pick the target column ----
    const int mBase = half * 8;                  // C/D layout: rows 0-7 / 8-15
    #pragma unroll
    for (int r = 0; r < 8; ++r) s_ori[wave][mBase + r][col] = d0[r];
    if (col < 4) {
      #pragma unroll
      for (int r = 0; r < 8; ++r) s_ori[wave][mBase + r][16 + col] = d1[r];
    }
    __syncthreads();

    const int  t     = __builtin_nontemporal_load(tgt + P0 + col);
    const bool valid = (t != IGNORE_IDX);
    const float ot   = s_ori[wave][col][valid ? t : 0];
    const float c    = logf(ot) - logf(total);   // log-prob of target class
    if (half == 0 && valid) { lsum -= c; lcnt += 1.f; }
    __syncthreads();
  }

  // ---- deterministic block reduction -> per-block partials ----
  s_red[tid] = lsum;
  __syncthreads();
  for (int s = THR2 / 2; s > 0; s >>= 1) {
    if (tid < s) s_red[tid] += s_red[tid + s];
    __syncthreads();
  }
  if (tid == 0) psum[blockIdx.x] = s_red[0];
  __syncthreads();
  s_red[tid] = lcnt;
  __syncthreads();
  for (int s = THR2 / 2; s > 0; s >>= 1) {
    if (tid < s) s_red[tid] += s_red[tid + s];
    __syncthreads();
  }
  if (tid == 0) pcnt[blockIdx.x] = s_red[0];
}

// ---------------------------------------------------------------------------
// Final reduction: sum 2048 block partials, divide by valid count.
// ---------------------------------------------------------------------------
__global__ void uce2d_finish_kernel(const float* __restrict__ psum,
                                    const float* __restrict__ pcnt,
                                    float* __restrict__ out) {
  __shared__ float ss[256], sc[256];
  float a = 0.f, c = 0.f;
  for (int i = threadIdx.x; i < BLK2; i += 256) { a += psum[i]; c += pcnt[i]; }
  ss[threadIdx.x] = a;
  sc[threadIdx.x] = c;
  __syncthreads();
  for (int s = 128; s > 0; s >>= 1) {
    if (threadIdx.x < s) {
      ss[threadIdx.x] += ss[threadIdx.x + s];
      sc[threadIdx.x] += sc[threadIdx.x + s];
    }
    __syncthreads();
  }
  if (threadIdx.x == 0) out[0] = ss[0] / fmaxf(sc[0], 1.f);
}

extern "C" void kernel_launch(void* const* d_in, const int* in_sizes, int n_in,
                              void* d_out, int out_size, void* d_ws, size_t ws_size,
                              hipStream_t stream) {
  (void)in_sizes; (void)n_in; (void)out_size; (void)ws_size;
  const float* in  = (const float*)d_in[0];   // [4,64,512,512] f32
  const int*   tgt = (const int*)d_in[1];     // [4,512,512] i32
  const int*   gid = (const int*)d_in[2];     // [64] i32
  float*       out = (float*)d_out;           // scalar loss

  float* w    = (float*)d_ws;
  float* psum = w;                            // [BLK2]
  float* pcnt = w + BLK2;                     // [BLK2]

  uce2d_main_kernel<<<BLK2, THR2, 0, stream>>>(in, tgt, gid, psum, pcnt);
  uce2d_finish_kernel<<<1, 256, 0, stream>>>(psum, pcnt, out);
}